// GCNCNN_15659450761590
// MI455X (gfx1250) — compile-verified
//
#include <hip/hip_runtime.h>
#include <hip/hip_bf16.h>

// Problem constants (fixed by the reference)
#define NN 50000   // nodes
#define TT 518     // timepoints
#define KK 32      // conv window
#define SS 3       // stride
#define LL 163     // (TT-KK)/SS + 1
#define HH 128     // hidden
#define EE 1600000 // edges
#define LPAD 164   // LL padded to multiple of 4 for WMMA K-chunks

typedef float v2f __attribute__((ext_vector_type(2)));
typedef float v8f __attribute__((ext_vector_type(8)));

// ---------------- kernel 0: zero the workspace (deg, srcw, vacc) -------------
__global__ void k_init(float* __restrict__ deg, float* __restrict__ srcw,
                       float* __restrict__ vacc) {
    int i = blockIdx.x * blockDim.x + threadIdx.x;
    if (i < NN) { deg[i] = 0.0f; srcw[i] = 0.0f; }
    if (i < LPAD) vacc[i] = 0.0f;
}

// ---------------- kernel 1: deg[col] += attr -------------------------------
__global__ void k_deg(const int* __restrict__ ei, const float* __restrict__ attr,
                      float* __restrict__ deg) {
    int e = blockIdx.x * blockDim.x + threadIdx.x;
    if (e < EE) atomicAdd(&deg[ei[EE + e]], attr[e]);
}

// ---------------- kernel 2: srcw[row] += dinv[row]*attr*dinv[col] ----------
__global__ void k_srcw(const int* __restrict__ ei, const float* __restrict__ attr,
                       const float* __restrict__ deg, float* __restrict__ srcw) {
    int e = blockIdx.x * blockDim.x + threadIdx.x;
    if (e < EE) {
        int r = ei[e], c = ei[EE + e];
        float nrm = __frsqrt_rn(deg[r] + 1.0f) * attr[e] * __frsqrt_rn(deg[c] + 1.0f);
        atomicAdd(&srcw[r], nrm);
    }
}

// ------- kernel 3: depthwise conv + ReLU, weighted-reduce into vacc[LL] ----
// Thread i owns node i. x is (T,N): column reads are coalesced across lanes.
__global__ void k_conv_accum(const float* __restrict__ x,
                             const float* __restrict__ conv_w,
                             const float* __restrict__ conv_b,
                             const float* __restrict__ deg,
                             const float* __restrict__ srcw,
                             float* __restrict__ vacc) {
    __shared__ float sv[LL];
    const int tid = threadIdx.x;
    for (int l = tid; l < LL; l += blockDim.x) sv[l] = 0.0f;
    __syncthreads();

    int i = blockIdx.x * blockDim.x + tid;
    const bool valid = (i < NN);
    const int ic = valid ? i : (NN - 1);      // clamp loads; weight 0 for tail

    float di   = __frsqrt_rn(deg[ic] + 1.0f);
    float wi   = valid ? (srcw[ic] + di * di) : 0.0f;
    float bias = conv_b[ic];

    // per-node conv weights: 32 floats, 128B aligned contiguous chunk
    float cw[KK];
    const float4* cw4 = reinterpret_cast<const float4*>(conv_w + (size_t)ic * KK);
#pragma unroll
    for (int q = 0; q < KK / 4; ++q) {
        float4 v = cw4[q];
        cw[4 * q + 0] = v.x; cw[4 * q + 1] = v.y;
        cw[4 * q + 2] = v.z; cw[4 * q + 3] = v.w;
    }

    const int lane = tid & 31;                // wave32
    for (int l = 0; l < LL; ++l) {
        const float* xp = x + (size_t)(l * SS) * NN + ic;
        float s = bias;
#pragma unroll
        for (int k = 0; k < KK; ++k) s = fmaf(cw[k], xp[(size_t)k * NN], s);
        float h = s > 0.0f ? s : 0.0f;        // ReLU
        float contrib = wi * h;
        // wave32 all-reduce (butterfly)
#pragma unroll
        for (int m = 16; m >= 1; m >>= 1) contrib += __shfl_xor(contrib, m, 32);
        if (lane == 0) atomicAdd(&sv[l], contrib);   // 8 LDS atomics per l per block
    }
    __syncthreads();
    for (int l = tid; l < LL; l += blockDim.x) atomicAdd(&vacc[l], sv[l]);
}

// ------- kernel 4: out = (vacc @ W)/N + b via V_WMMA_F32_16X16X4_F32 -------
// 1 block, 8 waves; wave w owns output columns [16w,16w+16). A uses row M=0 only.
// Branchless loop body: A via lane-varying vector load * per-lane {0,1} mask
// (mask/offset computed once), B via clamped-index load + zero-select.
// EXEC is all-ones and constant across every v_wmma.
__global__ void k_final_wmma(const float* __restrict__ vacc,
                             const float* __restrict__ W,
                             const float* __restrict__ b,
                             float* __restrict__ out) {
    const int lane = threadIdx.x & 31;
    const int wv   = threadIdx.x >> 5;                 // 0..7
    const int col  = (wv << 4) + (lane & 15);          // output column
    const int kb   = (lane < 16) ? 0 : 2;              // half-wave K offset (A and B)
    // A-matrix 16x4 f32 layout: lane m(0-15) holds row M=m, K=0/1 in VGPR0/1;
    // lanes 16-31 hold K=2/3. We only populate row M=0 -> lanes 0 and 16.
    const float amask = ((lane & 15) == 0) ? 1.0f : 0.0f;

    v8f c = {};
    for (int k0 = 0; k0 < LPAD; k0 += 4) {
        // A: one unconditional 8B load per lane (8B-aligned: k0%4==0, kb even)
        v2f av = *(const v2f*)(vacc + k0 + kb);
        v2f a;
        a[0] = av[0] * amask;
        a[1] = av[1] * amask;

        // B (4x16 f32): VGPR0 = row K=kb, VGPR1 = row K=kb+1.
        // Clamp row index so loads are unconditional; zero-select the pad row.
        int kr0 = k0 + kb, kr1 = kr0 + 1;
        int kc0 = (kr0 < LL) ? kr0 : (LL - 1);
        int kc1 = (kr1 < LL) ? kr1 : (LL - 1);
        float b0 = W[kc0 * HH + col];
        float b1 = W[kc1 * HH + col];
        v2f bm;
        bm[0] = (kr0 < LL) ? b0 : 0.0f;
        bm[1] = (kr1 < LL) ? b1 : 0.0f;

        // D = A*B + C  (exact fp32)
        c = __builtin_amdgcn_wmma_f32_16x16x4_f32(false, a, false, bm,
                                                  (short)0, c, false, false);
    }
    // D row M=0 lives in c[0] of lanes 0..15
    if (lane < 16) out[col] = c[0] * (1.0f / (float)NN) + b[col];
}

extern "C" void kernel_launch(void* const* d_in, const int* in_sizes, int n_in,
                              void* d_out, int out_size, void* d_ws, size_t ws_size,
                              hipStream_t stream) {
    const float* x      = (const float*)d_in[0];   // (T,N)
    const int*   ei     = (const int*)  d_in[1];   // (2,E)
    const float* attr   = (const float*)d_in[2];   // (E,)
    const float* conv_w = (const float*)d_in[3];   // (N,1,K)
    const float* conv_b = (const float*)d_in[4];   // (N,)
    const float* gcn_W  = (const float*)d_in[5];   // (L,H)
    const float* gcn_b  = (const float*)d_in[6];   // (H,)
    float* out = (float*)d_out;                    // (1,H) = 128 floats

    float* deg  = (float*)d_ws;        // N
    float* srcw = deg + NN;            // N
    float* vacc = srcw + NN;           // LPAD (zero-padded tail)

    const int B = 256;
    k_init      <<<(NN + B - 1) / B, B, 0, stream>>>(deg, srcw, vacc);
    k_deg       <<<(EE + B - 1) / B, B, 0, stream>>>(ei, attr, deg);
    k_srcw      <<<(EE + B - 1) / B, B, 0, stream>>>(ei, attr, deg, srcw);
    k_conv_accum<<<(NN + B - 1) / B, B, 0, stream>>>(x, conv_w, conv_b, deg, srcw, vacc);
    k_final_wmma<<<1, 256, 0, stream>>>(vacc, gcn_W, gcn_b, out);
}